// ProjectLayer_6468220748258
// MI455X (gfx1250) — compile-verified
//
#include <hip/hip_runtime.h>
#include <hip/hip_bf16.h>

typedef __attribute__((ext_vector_type(4))) float v4f;
typedef __attribute__((ext_vector_type(4))) int   v4i;

#define Bdim  4
#define Cdim  64
#define Hdim  256
#define Wdim  256
#define Ndim  65536
#define HWSZ  (Hdim * Wdim)          // 65536 elements per channel plane
#define NPT   4                      // n-values per thread
#define TPB   256                    // threads per block (8 wave32)
#define NTILE (TPB * NPT)            // 1024 n per block

// out[b][c][b2][n] = in[b][c][ pmap[b2][n][0] * W + pmap[b2][n][1] ]
__global__ __launch_bounds__(TPB) void ProjectLayer_gather_kernel(
    const float* __restrict__ in,    // (B, C, H, W) f32
    const int*   __restrict__ pmap,  // (B, N, 2)    i32
    float*       __restrict__ out)   // (B, C, B, N) f32
{
    __shared__ __align__(16) int tile[TPB * 2 * NPT];   // 1024 (row,col) pairs = 8 KB

    const int t  = threadIdx.x;
    const int bx = blockIdx.x;   // n-tile index
    const int b2 = blockIdx.y;   // index-map batch
    const int b  = blockIdx.z;   // feature batch

    const int n0 = bx * NTILE + t * NPT;

    // ---- stage this block's 1024 index pairs into LDS via the CDNA5 async path ----
    // Each lane async-loads its own 32 bytes (4 int2 pairs) into its own LDS slot.
    // GVS addressing: saddr (SGPR pair) + per-lane 32-bit voffset; the instruction
    // offset applies to both the global and LDS address, so two B128s cover 32 B.
    const int* gbase = pmap + ((size_t)b2 * Ndim + (size_t)bx * NTILE) * 2;
    unsigned lds_addr = (unsigned)(uintptr_t)(&tile[0]) + (unsigned)(t * 32);
    unsigned goff     = (unsigned)(t * 32);
    asm volatile(
        "global_load_async_to_lds_b128 %0, %1, %2 offset:0\n\t"
        "global_load_async_to_lds_b128 %0, %1, %2 offset:16\n\t"
        "s_wait_asynccnt 0"
        :
        : "v"(lds_addr), "v"(goff), "s"(gbase)
        : "memory");
    // ASYNCcnt is per-wave and each lane reads back only its own slot -> no barrier.

    const v4i p01 = *(const v4i*)&tile[t * 8 + 0];   // (r0,c0,r1,c1)
    const v4i p23 = *(const v4i*)&tile[t * 8 + 4];   // (r2,c2,r3,c3)
    const int pix0 = p01.x * Wdim + p01.y;
    const int pix1 = p01.z * Wdim + p01.w;
    const int pix2 = p23.x * Wdim + p23.y;
    const int pix3 = p23.z * Wdim + p23.w;

    // plane pointer walks the 64 channels of feature-batch b (uniform stride -> SGPR math)
    const float* plane = in + (size_t)b * Cdim * HWSZ;
    // out[((b*C + c)*B + b2)*N + n0], c stride = B*N elements
    float* op = out + (((size_t)(b * Cdim) * Bdim) + (size_t)b2) * (size_t)Ndim + (size_t)n0;

    #pragma unroll 4
    for (int c = 0; c < Cdim; ++c) {
        v4f v;
        v.x = plane[pix0];           // scattered 4B reads; whole input (64 MiB) is L2-resident
        v.y = plane[pix1];
        v.z = plane[pix2];
        v.w = plane[pix3];
        // 16B coalesced non-temporal store: don't pollute L2 with the 256 MiB output stream
        __builtin_nontemporal_store(v, (v4f*)op);
        plane += HWSZ;
        op    += (size_t)Bdim * Ndim;
    }
}

extern "C" void kernel_launch(void* const* d_in, const int* in_sizes, int n_in,
                              void* d_out, int out_size, void* d_ws, size_t ws_size,
                              hipStream_t stream) {
    const float* in   = (const float*)d_in[0];   // (B, C, H, W) float32
    const int*   pmap = (const int*)d_in[1];     // (B, N, 2)    int32
    float*       out  = (float*)d_out;           // (B, C, B, N) float32

    dim3 grid(Ndim / NTILE, Bdim, Bdim);         // (64, 4, 4) = 1024 blocks
    dim3 block(TPB);
    ProjectLayer_gather_kernel<<<grid, block, 0, stream>>>(in, pmap, out);
}